// PaddedMasking_9491877724175
// MI455X (gfx1250) — compile-verified
//
#include <hip/hip_runtime.h>
#include <stdint.h>

// ---------------------------------------------------------------------------
// PaddedMasking: out = x * mask(h, w)
//   x: (64, 3, 384, 384) fp32, patch_vals: (21, 21) int32 in {0,1}
//   mask(h,w) = patch_vals[r,c] if (h,w) inside the 12x12 patch at
//               (6 + r*18, 6 + c*18), else 1.0
// Bandwidth-bound: 226 MB @ 23.3 TB/s => ~9.7 us floor.
// Hot loop: 8-deep 128-bit non-temporal load/store pipeline per thread.
// Prologue: CDNA5 async Global->LDS DMA (ASYNCcnt) stages the 21x21 LUT.
// ---------------------------------------------------------------------------

typedef float v4f __attribute__((ext_vector_type(4)));

#define IMG_H   384
#define IMG_W   384
#define PAD     6
#define STRIDE  18
#define PATCH   12
#define NP      21
#define PLANES  192                 // B*C = 64*3
#define W4      (IMG_W / 4)         // 96 float4 per row
#define PLANE4  (IMG_H * IMG_W / 4) // 36864 float4 per plane
#define PPT     8                   // planes per thread
#define NGROUPS (PLANES / PPT)      // 24
#define BLOCK   256
#define NPV     (NP * NP)           // 441 dwords = 1764 B

__global__ __launch_bounds__(BLOCK)
void padded_mask_kernel(const float* __restrict__ x,
                        const int*   __restrict__ pv,
                        float*       __restrict__ out) {
    // LUT staged raw (int32) into LDS via async Global->LDS DMA.
    // 444 dwords so the b128 tail stays in-bounds on the LDS side.
    __shared__ int s_pv[444];

    {
        const int t = threadIdx.x;
        // Low 32 bits of a generic pointer to __shared__ == LDS byte offset
        // (aperture scheme).
        const unsigned lds_base = (unsigned)(uintptr_t)(void*)&s_pv[0];
        if (t < 110) {
            // Lanes 0..109: 16 B each -> dwords 0..439.
            const unsigned           ldso = lds_base + (unsigned)t * 16u;
            const unsigned long long ga   = (unsigned long long)(uintptr_t)pv
                                          + (unsigned long long)t * 16ull;
            asm volatile("global_load_async_to_lds_b128 %0, %1, off"
                         :: "v"(ldso), "v"(ga) : "memory");
        } else if (t == 110) {
            // Final dword 440.
            const unsigned           ldso = lds_base + 1760u;
            const unsigned long long ga   = (unsigned long long)(uintptr_t)pv + 1760ull;
            asm volatile("global_load_async_to_lds_b32 %0, %1, off"
                         :: "v"(ldso), "v"(ga) : "memory");
        }
        // Each wave drains its own ASYNCcnt, then the barrier makes the LDS
        // contents visible to the whole block.
        asm volatile("s_wait_asynccnt 0x0" ::: "memory");
    }
    __syncthreads();

    const int tid = blockIdx.x * BLOCK + threadIdx.x;
    const int pos = tid % PLANE4;   // float4 index within one (H,W) plane
    const int grp = tid / PLANE4;   // which group of PPT planes

    const int h  = pos / W4;
    const int w4 = pos - h * W4;

    // Row patch index (or -1 if this row is padding).
    int r = -1;
    {
        const int rel = h - PAD;
        if ((unsigned)rel < (unsigned)(NP * STRIDE)) {
            const int q = rel / STRIDE;             // const-div -> mulhi
            if (rel - q * STRIDE < PATCH) r = q;
        }
    }

    // 4-wide mask for the float4 this thread owns; reused across PPT planes.
    v4f m;
#pragma unroll
    for (int k = 0; k < 4; ++k) {
        const int rel = w4 * 4 + k - PAD;
        int c = -1;
        if ((unsigned)rel < (unsigned)(NP * STRIDE)) {
            const int q = rel / STRIDE;
            if (rel - q * STRIDE < PATCH) c = q;
        }
        const bool inside = (r >= 0) & (c >= 0);
        const int  idx    = inside ? (r * NP + c) : 0;
        m[k] = inside ? (float)s_pv[idx] : 1.0f;    // ds_load + v_cvt + cndmask
    }

    const size_t base = (size_t)grp * PPT * PLANE4 + (size_t)pos;
    const v4f* __restrict__ src = (const v4f*)x + base;
    v4f*       __restrict__ dst = (v4f*)out + base;

    // Issue all PPT b128 loads first (8-deep MLP), non-temporal since the
    // 226 MB stream is touched once and exceeds the 192 MB L2.
    v4f v[PPT];
#pragma unroll
    for (int p = 0; p < PPT; ++p)
        v[p] = __builtin_nontemporal_load(src + (size_t)p * PLANE4);
#pragma unroll
    for (int p = 0; p < PPT; ++p) {
        v4f t = v[p] * m;
        __builtin_nontemporal_store(t, dst + (size_t)p * PLANE4);
    }
}

extern "C" void kernel_launch(void* const* d_in, const int* in_sizes, int n_in,
                              void* d_out, int out_size, void* d_ws, size_t ws_size,
                              hipStream_t stream) {
    const float* x   = (const float*)d_in[0];  // (64,3,384,384) fp32
    const int*   pv  = (const int*)d_in[1];    // (21,21) int32
    float*       out = (float*)d_out;

    // 36864 positions * 24 plane-groups = 884736 threads = 3456 blocks, exact.
    const int total  = PLANE4 * NGROUPS;
    const int blocks = total / BLOCK;
    padded_mask_kernel<<<blocks, BLOCK, 0, stream>>>(x, pv, out);
}